// sage_10892037063085
// MI455X (gfx1250) — compile-verified
//
#include <hip/hip_runtime.h>

typedef float v2f __attribute__((ext_vector_type(2)));
typedef float v8f __attribute__((ext_vector_type(8)));

#define NNODES 25600
#define NEDGES 409600
#define ACT_RELU 1
#define ACT_LEAKY 2

#define FINAL_K 51200
#define KSPLIT 64
#define KCHUNK (FINAL_K / KSPLIT)   // 800, divisible by 8
#define NPAD 112                    // 100 cols padded to 7*16

constexpr int ilog2c(int v) { int s = 0; while ((1 << s) < v) ++s; return s; }

// ---------------- degree ----------------
__global__ __launch_bounds__(256) void deg_count(const int* __restrict__ dst,
                                                 float* __restrict__ deg) {
    int e = blockIdx.x * 256 + threadIdx.x;              // exactly NEDGES threads
    atomicAdd(&deg[dst[e]], 1.0f);
}

__global__ __launch_bounds__(256) void deg_invert(float* __restrict__ deg) {
    int n = blockIdx.x * 256 + threadIdx.x;              // exactly NNODES threads
    deg[n] = 1.0f / fmaxf(deg[n], 1.0f);
}

// ---------------- mean-aggregation scatter (sum part) ----------------
// one thread per (edge, 4-channel chunk): one b128 load + 4 f32 atomics.
__global__ __launch_bounds__(256) void scatter_add(const float* __restrict__ x,
                                                   const int* __restrict__ src,
                                                   const int* __restrict__ dst,
                                                   float* __restrict__ agg,
                                                   unsigned c4mask, int dshift) {
    unsigned idx = blockIdx.x * 256u + threadIdx.x;
    unsigned e  = idx >> (dshift - 2);
    unsigned c  = (idx & c4mask) << 2;
    const float4 v = *(const float4*)(x + ((unsigned)src[e] << dshift) + c);
    float* p = agg + ((unsigned)dst[e] << dshift) + c;
    atomicAdd(p + 0, v.x);
    atomicAdd(p + 1, v.y);
    atomicAdd(p + 2, v.z);
    atomicAdd(p + 3, v.w);
}

// ---------------- fused SAGEConv GEMM ----------------
// out = (agg * invdeg) @ Wl^T + x @ Wr^T + bl, + activation.
// Block = 8 waves sharing one 16-row m-tile staged in LDS (XOR-swizzled,
// bank-conflict-free b64 reads); each wave owns one 16-col n-tile and runs
// two independent v_wmma_f32_16x16x4_f32 accumulator chains.
// Uniform k-loop counter (no per-lane exec masking), compile-time shapes.
template <int DIN, int DOUT, int ACT>
__global__ __launch_bounds__(256) void sage_gemm(const float* __restrict__ agg,
                                                 const float* __restrict__ invdeg,
                                                 const float* __restrict__ xin,
                                                 const float* __restrict__ Wl,
                                                 const float* __restrict__ bl,
                                                 const float* __restrict__ Wr,
                                                 float* __restrict__ out) {
    constexpr int DSH     = ilog2c(DIN);
    constexpr int TILES_N = DOUT / 16;
    constexpr int NGROUPS = (TILES_N + 7) / 8;
    constexpr int TOTAL   = 16 * DIN;           // elements per staged matrix
    __shared__ float lds[2 * TOTAL];            // <= 64 KB for DIN=512

    const int tid = threadIdx.x;
    const int mt  = (int)blockIdx.x / NGROUPS;
    const int ng  = (int)blockIdx.x % NGROUPS;
    const int m0  = mt << 4;

    // cooperative, coalesced fill of scaled-agg tile and x tile
#pragma unroll
    for (int i = 0; i < TOTAL / 256; ++i) {
        int idx = i * 256 + tid;
        int rr  = idx >> DSH;
        int cc  = idx & (DIN - 1);
        int lo  = (rr << DSH) + (cc ^ ((rr << 2) & (DIN - 1)));   // XOR swizzle
        size_t g = ((size_t)(m0 + rr) << DSH) + cc;
        lds[lo]         = agg[g] * invdeg[m0 + rr];
        lds[TOTAL + lo] = xin[g];
    }
    __syncthreads();

    const int wave = tid >> 5;
    const int nt   = ng * 8 + wave;             // wave-uniform
    if (nt < TILES_N) {                         // EXEC stays all-ones per wave
        const int lane = tid & 31;
        const int r    = lane & 15;             // A row / B,D col
        const int khi2 = (lane >> 4) << 1;      // k sub-offset (0 or 2)
        const int n    = nt * 16 + r;
        const float* __restrict__ la  = lds + (r << DSH);
        const float* __restrict__ lx  = la + TOTAL;
        const float* __restrict__ b0r = Wl + (size_t)n * DIN;  // B[k][n]=W[n*DIN+k]
        const float* __restrict__ b1r = Wr + (size_t)n * DIN;
        const int sw = (r << 2) & (DIN - 1);

        v8f acc0 = {}, acc1 = {};
#pragma unroll 4
        for (int kb = 0; kb < DIN; kb += 4) {   // uniform trip count
            const int k  = kb + khi2;
            const int ks = k ^ sw;              // sw % 4 == 0 -> pair contiguous
            v2f a0, a1, b0, b1;
            a0.x = la[ks];  a0.y = la[ks + 1];
            a1.x = lx[ks];  a1.y = lx[ks + 1];
            b0.x = b0r[k];  b0.y = b0r[k + 1];
            b1.x = b1r[k];  b1.y = b1r[k + 1];
            acc0 = __builtin_amdgcn_wmma_f32_16x16x4_f32(
                false, a0, false, b0, (short)0, acc0, false, false);
            acc1 = __builtin_amdgcn_wmma_f32_16x16x4_f32(
                false, a1, false, b1, (short)0, acc1, false, false);
        }
        const float bias = bl[n];
#pragma unroll
        for (int j = 0; j < 8; ++j) {
            float v = acc0[j] + acc1[j] + bias;          // D[j + 8*khi][r]
            if constexpr (ACT == ACT_RELU) v = fmaxf(v, 0.0f);
            else                           v = (v > 0.0f) ? v : 0.1f * v;
            out[(size_t)(m0 + j + (khi2 << 2)) * DOUT + n] = v;
        }
    }
}

// ---------------- readout GEMM: [256,51200] @ Wn^T -> [256,100] ----------------
// split-K WMMA, dual accumulator chains, fp32 atomic reduction into [256,112].
__global__ __launch_bounds__(256) void final_gemm(const float* __restrict__ h,
                                                  const float* __restrict__ Wn,
                                                  float* __restrict__ accum) {
    const int lane = threadIdx.x & 31;
    const int wave = threadIdx.x >> 5;
    int wid = blockIdx.x * 8 + wave;            // 16*7*KSPLIT waves total
    const int ks = wid % KSPLIT; wid /= KSPLIT;
    const int tn = wid % 7;
    const int tm = wid / 7;
    const int r    = lane & 15;
    const int khi2 = (lane >> 4) << 1;
    const int m = tm * 16 + r;
    const int n = tn * 16 + r;
    const int ncl = (n < 100) ? n : 99;         // clamp OOB weight rows;
                                                // accum cols 100..111 never read
    const float* __restrict__ arow = h  + (size_t)m   * FINAL_K;
    const float* __restrict__ brow = Wn + (size_t)ncl * FINAL_K;

    v8f acc0 = {}, acc1 = {};
    const int k0 = ks * KCHUNK;
#pragma unroll 2
    for (int kb = 0; kb < KCHUNK; kb += 8) {    // uniform trip count
        const int k = k0 + kb + khi2;
        v2f a0, a1, b0, b1;
        a0.x = arow[k];     a0.y = arow[k + 1];
        b0.x = brow[k];     b0.y = brow[k + 1];
        a1.x = arow[k + 4]; a1.y = arow[k + 5];
        b1.x = brow[k + 4]; b1.y = brow[k + 5];
        acc0 = __builtin_amdgcn_wmma_f32_16x16x4_f32(
            false, a0, false, b0, (short)0, acc0, false, false);
        acc1 = __builtin_amdgcn_wmma_f32_16x16x4_f32(
            false, a1, false, b1, (short)0, acc1, false, false);
    }
#pragma unroll
    for (int j = 0; j < 8; ++j)
        atomicAdd(&accum[(size_t)(tm * 16 + j + (khi2 << 2)) * NPAD + n],
                  acc0[j] + acc1[j]);
}

__global__ __launch_bounds__(256) void final_bias_relu(const float* __restrict__ accum,
                                                       const float* __restrict__ bn,
                                                       float* __restrict__ out) {
    int i = blockIdx.x * 256 + threadIdx.x;     // exactly 256*100 threads
    int mrow = i / 100, c = i - mrow * 100;
    out[i] = fmaxf(accum[(size_t)mrow * NPAD + c] + bn[c], 0.0f);
}

// ---------------- host side ----------------
template <int DIN, int DOUT, int ACT>
static void run_conv(const float* xin,
                     const float* Wl, const float* bl, const float* Wr,
                     float* out, float* agg, const float* invdeg,
                     const int* src, const int* dst, hipStream_t stream) {
    hipMemsetAsync(agg, 0, (size_t)NNODES * DIN * sizeof(float), stream);
    constexpr int DSH = ilog2c(DIN);
    constexpr unsigned TOTAL = (unsigned)NEDGES << (DSH - 2);   // edges * DIN/4
    scatter_add<<<TOTAL / 256u, 256, 0, stream>>>(xin, src, dst, agg,
                                                  (unsigned)(DIN / 4 - 1), DSH);
    constexpr int NGROUPS = (DOUT / 16 + 7) / 8;                // 8 n-tiles / block
    sage_gemm<DIN, DOUT, ACT><<<(NNODES / 16) * NGROUPS, 256, 0, stream>>>(
        agg, invdeg, xin, Wl, bl, Wr, out);
}

extern "C" void kernel_launch(void* const* d_in, const int* in_sizes, int n_in,
                              void* d_out, int out_size, void* d_ws, size_t ws_size,
                              hipStream_t stream) {
    const float* x  = (const float*)d_in[0];
    const int*  ei  = (const int*)d_in[1];
    const int*  src = ei;                                    // edge_index[0]
    const int*  dst = ei + NEDGES;                           // edge_index[1]
    const float* Wl[6], *bl[6], *Wr[6];
    for (int i = 0; i < 6; ++i) {
        Wl[i] = (const float*)d_in[2 + 3 * i];
        bl[i] = (const float*)d_in[3 + 3 * i];
        Wr[i] = (const float*)d_in[4 + 3 * i];
    }
    const float* Wn = (const float*)d_in[20];
    const float* bn = (const float*)d_in[21];

    float* w      = (float*)d_ws;
    float* invdeg = w;                                       // NNODES
    float* agg    = invdeg + NNODES;                         // NNODES*512 max
    float* hA     = agg + (size_t)NNODES * 512;              // NNODES*512
    float* hB     = hA  + (size_t)NNODES * 512;              // NNODES*512
    float* accum  = hB  + (size_t)NNODES * 512;              // 256*112

    // degree -> 1/max(deg,1), recomputed every call (deterministic)
    hipMemsetAsync(invdeg, 0, NNODES * sizeof(float), stream);
    deg_count<<<NEDGES / 256, 256, 0, stream>>>(dst, invdeg);
    deg_invert<<<NNODES / 256, 256, 0, stream>>>(invdeg);

    // block 1: conv(32->64) relu, conv(64->64) leaky
    run_conv<32, 64, ACT_RELU >(x,  Wl[0], bl[0], Wr[0], hA, agg, invdeg, src, dst, stream);
    run_conv<64, 64, ACT_LEAKY>(hA, Wl[1], bl[1], Wr[1], hB, agg, invdeg, src, dst, stream);
    // block 2: conv(64->256) relu, conv(256->256) leaky
    run_conv<64,  256, ACT_RELU >(hB, Wl[2], bl[2], Wr[2], hA, agg, invdeg, src, dst, stream);
    run_conv<256, 256, ACT_LEAKY>(hA, Wl[3], bl[3], Wr[3], hB, agg, invdeg, src, dst, stream);
    // block 3: conv(256->512) relu, conv(512->512) leaky
    run_conv<256, 512, ACT_RELU >(hB, Wl[4], bl[4], Wr[4], hA, agg, invdeg, src, dst, stream);
    run_conv<512, 512, ACT_LEAKY>(hA, Wl[5], bl[5], Wr[5], hB, agg, invdeg, src, dst, stream);

    // readout: [256,51200] @ Wn^T + bn, relu
    hipMemsetAsync(accum, 0, 256 * NPAD * sizeof(float), stream);
    final_gemm<<<(16 * 7 * KSPLIT) / 8, 256, 0, stream>>>(hB, Wn, accum);
    final_bias_relu<<<(256 * 100) / 256, 256, 0, stream>>>(accum, bn, (float*)d_out);
}